// SimpleMambaBlock_84937273246163
// MI455X (gfx1250) — compile-verified
//
#include <hip/hip_runtime.h>
#include <hip/hip_bf16.h>

typedef float v2f __attribute__((ext_vector_type(2)));
typedef float v8f __attribute__((ext_vector_type(8)));

constexpr int kDim = 384;     // DIM
constexpr int kDst = 16;      // D_STATE
constexpr int kL   = 4096;    // H*W
constexpr int kB   = 4;       // batch
constexpr int kP   = 800;     // 2*DIM + 2*D_STATE
constexpr int kM   = kB * kL; // 16384 rows

constexpr int TS = 72;        // LDS tile stride (2*72 % 64 == 16 -> halves hit disjoint banks)

__device__ __forceinline__ float softplus_f(float v) {
    return (v > 20.f) ? v : log1pf(__expf(v));
}

// ---------------------------------------------------------------------------
// Kernel 1: proj = seq @ W_proj^T + b_proj, fused split + softplus epilogue.
// A[m][k] = x[b][k][l]  (m = b*4096 + l)  -> columns of A contiguous in memory
// B[k][n] = W_proj[n][k]
// ---------------------------------------------------------------------------
__global__ __launch_bounds__(256)
void proj_gemm_kernel(const float* __restrict__ x,      // [B][DIM][L]
                      const float* __restrict__ Wp,     // [P][DIM]
                      const float* __restrict__ bp,     // [P]
                      const float* __restrict__ dtb,    // [DIM]
                      float* __restrict__ xin,          // [M][DIM]
                      float* __restrict__ dlt,          // [M][DIM]
                      float* __restrict__ Bws,          // [M][16]
                      float* __restrict__ Cws)          // [M][16]
{
    __shared__ float sA[32 * TS];   // [k][m], padded
    __shared__ float sB[32 * TS];   // [k][n], padded

    const int tid  = threadIdx.x;
    const int wave = tid >> 5;
    const int lane = tid & 31;
    const int half = lane >> 4;
    const int l16  = lane & 15;

    const int m0 = blockIdx.x * 64;
    const int n0 = blockIdx.y * 64;

    const int wm = (wave & 1) * 32;   // wave's M offset (two 16-row subtiles)
    const int wn = (wave >> 1) * 16;  // wave's N offset

    v8f acc0 = {};
    v8f acc1 = {};

    const int bidx = m0 >> 12;        // 4096 rows per batch, tiles never straddle
    const int l0   = m0 & 4095;
    const float* xb = x + (size_t)bidx * kDim * kL;

    for (int k0 = 0; k0 < kDim; k0 += 32) {
        // A tile: 32 k-rows x 64 m-cols. Column-contiguous global loads.
        {
            const int j  = tid >> 3;          // k within tile: 0..31
            const int i0 = (tid & 7) * 8;     // m within tile: 0..56
            const float* src = xb + (size_t)(k0 + j) * kL + l0 + i0;
            float4 v0 = *(const float4*)(src);
            float4 v1 = *(const float4*)(src + 4);
            *(float4*)(&sA[j * TS + i0])     = v0;
            *(float4*)(&sA[j * TS + i0 + 4]) = v1;
        }
        // B tile: B[k][n] = Wp[n][k]; rows of Wp contiguous in k.
        {
            const int nn = tid >> 2;          // n within tile: 0..63
            const int j0 = (tid & 3) * 8;     // k within tile: 0,8,16,24
            const int ng = n0 + nn;
            float vv[8];
            if (ng < kP) {
                const float* src = Wp + (size_t)ng * kDim + k0 + j0;
                float4 v0 = *(const float4*)(src);
                float4 v1 = *(const float4*)(src + 4);
                vv[0]=v0.x; vv[1]=v0.y; vv[2]=v0.z; vv[3]=v0.w;
                vv[4]=v1.x; vv[5]=v1.y; vv[6]=v1.z; vv[7]=v1.w;
            } else {
                #pragma unroll
                for (int e = 0; e < 8; ++e) vv[e] = 0.f;
            }
            #pragma unroll
            for (int e = 0; e < 8; ++e) sB[(j0 + e) * TS + nn] = vv[e];
        }
        __syncthreads();

        #pragma unroll
        for (int kk = 0; kk < 32; kk += 4) {
            const int kr = kk + 2 * half;
            v2f a0, a1, bf;
            a0.x = sA[kr * TS + wm + l16];
            a0.y = sA[(kr + 1) * TS + wm + l16];
            a1.x = sA[kr * TS + wm + 16 + l16];
            a1.y = sA[(kr + 1) * TS + wm + 16 + l16];
            bf.x = sB[kr * TS + wn + l16];
            bf.y = sB[(kr + 1) * TS + wn + l16];
            acc0 = __builtin_amdgcn_wmma_f32_16x16x4_f32(false, a0, false, bf, (short)0, acc0, false, false);
            acc1 = __builtin_amdgcn_wmma_f32_16x16x4_f32(false, a1, false, bf, (short)0, acc1, false, false);
        }
        __syncthreads();
    }

    // Epilogue: split P columns -> x_in / softplus(delta) / B / C
    const int ng = n0 + wn + l16;
    if (ng < kP) {
        const float bias = bp[ng];
        #pragma unroll
        for (int r = 0; r < 8; ++r) {
            const int mr0 = m0 + wm + r + 8 * half;
            const int mr1 = mr0 + 16;
            float v0 = acc0[r] + bias;
            float v1 = acc1[r] + bias;
            if (ng < kDim) {
                xin[(size_t)mr0 * kDim + ng] = v0;
                xin[(size_t)mr1 * kDim + ng] = v1;
            } else if (ng < 2 * kDim) {
                const int d = ng - kDim;
                const float db = dtb[d];
                dlt[(size_t)mr0 * kDim + d] = softplus_f(v0 + db);
                dlt[(size_t)mr1 * kDim + d] = softplus_f(v1 + db);
            } else if (ng < 2 * kDim + kDst) {
                const int s = ng - 2 * kDim;
                Bws[(size_t)mr0 * kDst + s] = v0;
                Bws[(size_t)mr1 * kDst + s] = v1;
            } else {
                const int s = ng - 2 * kDim - kDst;
                Cws[(size_t)mr0 * kDst + s] = v0;
                Cws[(size_t)mr1 * kDst + s] = v1;
            }
        }
    }
}

// ---------------------------------------------------------------------------
// Kernel 2: selective scan. lane = (dim_local, n); 16 dims x 16 states/block.
// Double-buffered LDS chunks of 32 timesteps hide L2 latency.
// ---------------------------------------------------------------------------
__global__ __launch_bounds__(256)
void scan_kernel(const float* __restrict__ xin,   // [M][DIM]
                 const float* __restrict__ dlt,   // [M][DIM]
                 const float* __restrict__ Bws,   // [M][16]
                 const float* __restrict__ Cws,   // [M][16]
                 const float* __restrict__ A_log, // [DIM][16]
                 const float* __restrict__ Dp,    // [DIM]
                 float* __restrict__ yout)        // [M][DIM]
{
    __shared__ float sD[2][32 * 16];
    __shared__ float sX[2][32 * 16];
    __shared__ float sBv[2][32 * 16];
    __shared__ float sCv[2][32 * 16];

    const int tid = threadIdx.x;
    const int n   = tid & 15;
    const int dl  = tid >> 4;                 // 0..15
    const int blk = blockIdx.x;               // 0..95
    const int b   = blk / (kDim / 16);
    const int d0  = (blk % (kDim / 16)) * 16;
    const int d   = d0 + dl;

    const float A_dn = -__expf(A_log[d * kDst + n]);
    const float D_d  = Dp[d];

    const float* xin_b = xin + (size_t)b * kL * kDim;
    const float* dlt_b = dlt + (size_t)b * kL * kDim;
    const float* Bb    = Bws + (size_t)b * kL * kDst;
    const float* Cb    = Cws + (size_t)b * kL * kDst;
    float*       yb    = yout + (size_t)b * kL * kDim;

    auto load_chunk = [&](int t0, int buf) {
        #pragma unroll
        for (int e = tid; e < 512; e += 256) {
            const int row = e >> 4, col = e & 15;
            const size_t go = (size_t)(t0 + row) * kDim + d0 + col;
            sD[buf][e] = dlt_b[go];
            sX[buf][e] = xin_b[go];
            const size_t gb = (size_t)t0 * kDst + e;   // B/C chunks are contiguous
            sBv[buf][e] = Bb[gb];
            sCv[buf][e] = Cb[gb];
        }
    };

    load_chunk(0, 0);
    __syncthreads();

    float h = 0.f;
    constexpr int NCHUNK = kL / 32;   // 128
    for (int c = 0; c < NCHUNK; ++c) {
        const int buf = c & 1;
        if (c + 1 < NCHUNK) load_chunk((c + 1) * 32, buf ^ 1);

        const int tbase = c * 32;
        #pragma unroll 8
        for (int tt = 0; tt < 32; ++tt) {
            const float dt = sD[buf][tt * 16 + dl];   // broadcast over 16 lanes
            const float xv = sX[buf][tt * 16 + dl];
            const float bn = sBv[buf][tt * 16 + n];
            const float cn = sCv[buf][tt * 16 + n];
            h = __expf(dt * A_dn) * h + (dt * xv) * bn;
            float yp = h * cn;
            yp += __shfl_xor(yp, 1);
            yp += __shfl_xor(yp, 2);
            yp += __shfl_xor(yp, 4);
            yp += __shfl_xor(yp, 8);
            if (n == 0)
                yb[(size_t)(tbase + tt) * kDim + d] = yp + xv * D_d;
        }
        __syncthreads();
    }
}

// ---------------------------------------------------------------------------
// Kernel 3: out = y @ W_out^T + b_out, transposed store via LDS so the
// [B][C][L] layout is written with coalesced float4s.
// ---------------------------------------------------------------------------
__global__ __launch_bounds__(256)
void out_gemm_kernel(const float* __restrict__ y,   // [M][DIM]
                     const float* __restrict__ Wo,  // [DIM][DIM]
                     const float* __restrict__ bo,  // [DIM]
                     float* __restrict__ out)       // [B][DIM][L]
{
    __shared__ float smem[64 * TS];                 // union: (sA|sB) then sC
    float* sA = smem;                               // 32*TS
    float* sB = smem + 32 * TS;                     // 32*TS
    float* sC = smem;                               // 64*TS after final barrier

    const int tid  = threadIdx.x;
    const int wave = tid >> 5;
    const int lane = tid & 31;
    const int half = lane >> 4;
    const int l16  = lane & 15;

    const int m0 = blockIdx.x * 64;
    const int n0 = blockIdx.y * 64;
    const int wm = (wave & 1) * 32;
    const int wn = (wave >> 1) * 16;

    v8f acc0 = {};
    v8f acc1 = {};

    for (int k0 = 0; k0 < kDim; k0 += 32) {
        // A tile: A[m][k] = y[m][k], row-major -> k contiguous
        {
            const int i  = tid >> 2;          // m: 0..63
            const int j0 = (tid & 3) * 8;     // k: 0,8,16,24
            const float* src = y + (size_t)(m0 + i) * kDim + k0 + j0;
            float4 v0 = *(const float4*)(src);
            float4 v1 = *(const float4*)(src + 4);
            sA[(j0 + 0) * TS + i] = v0.x; sA[(j0 + 1) * TS + i] = v0.y;
            sA[(j0 + 2) * TS + i] = v0.z; sA[(j0 + 3) * TS + i] = v0.w;
            sA[(j0 + 4) * TS + i] = v1.x; sA[(j0 + 5) * TS + i] = v1.y;
            sA[(j0 + 6) * TS + i] = v1.z; sA[(j0 + 7) * TS + i] = v1.w;
        }
        // B tile: B[k][n] = Wo[n][k]
        {
            const int nn = tid >> 2;
            const int j0 = (tid & 3) * 8;
            const float* src = Wo + (size_t)(n0 + nn) * kDim + k0 + j0;
            float4 v0 = *(const float4*)(src);
            float4 v1 = *(const float4*)(src + 4);
            sB[(j0 + 0) * TS + nn] = v0.x; sB[(j0 + 1) * TS + nn] = v0.y;
            sB[(j0 + 2) * TS + nn] = v0.z; sB[(j0 + 3) * TS + nn] = v0.w;
            sB[(j0 + 4) * TS + nn] = v1.x; sB[(j0 + 5) * TS + nn] = v1.y;
            sB[(j0 + 6) * TS + nn] = v1.z; sB[(j0 + 7) * TS + nn] = v1.w;
        }
        __syncthreads();

        #pragma unroll
        for (int kk = 0; kk < 32; kk += 4) {
            const int kr = kk + 2 * half;
            v2f a0, a1, bf;
            a0.x = sA[kr * TS + wm + l16];
            a0.y = sA[(kr + 1) * TS + wm + l16];
            a1.x = sA[kr * TS + wm + 16 + l16];
            a1.y = sA[(kr + 1) * TS + wm + 16 + l16];
            bf.x = sB[kr * TS + wn + l16];
            bf.y = sB[(kr + 1) * TS + wn + l16];
            acc0 = __builtin_amdgcn_wmma_f32_16x16x4_f32(false, a0, false, bf, (short)0, acc0, false, false);
            acc1 = __builtin_amdgcn_wmma_f32_16x16x4_f32(false, a1, false, bf, (short)0, acc1, false, false);
        }
        __syncthreads();
    }

    // Stage D tile into LDS as [n][m] (stride TS), add bias
    const float bias = bo[n0 + wn + l16];
    #pragma unroll
    for (int r = 0; r < 8; ++r) {
        const int mm = wm + r + 8 * half;
        sC[(wn + l16) * TS + mm]      = acc0[r] + bias;
        sC[(wn + l16) * TS + mm + 16] = acc1[r] + bias;
    }
    __syncthreads();

    // Coalesced store: out[b][n][l] rows are contiguous in l
    const int bidx = m0 >> 12;
    const int l0   = m0 & 4095;
    const int nn   = tid >> 2;            // 0..63
    const int ms   = (tid & 3) * 16;      // 0,16,32,48
    float* dst = out + ((size_t)bidx * kDim + n0 + nn) * kL + l0 + ms;
    #pragma unroll
    for (int e = 0; e < 4; ++e) {
        float4 v;
        v.x = sC[nn * TS + ms + e * 4 + 0];
        v.y = sC[nn * TS + ms + e * 4 + 1];
        v.z = sC[nn * TS + ms + e * 4 + 2];
        v.w = sC[nn * TS + ms + e * 4 + 3];
        *(float4*)(dst + e * 4) = v;
    }
}

// ---------------------------------------------------------------------------
extern "C" void kernel_launch(void* const* d_in, const int* in_sizes, int n_in,
                              void* d_out, int out_size, void* d_ws, size_t ws_size,
                              hipStream_t stream) {
    const float* x     = (const float*)d_in[0];
    const float* Wp    = (const float*)d_in[1];
    const float* bp    = (const float*)d_in[2];
    const float* A_log = (const float*)d_in[3];
    const float* Dp    = (const float*)d_in[4];
    const float* dtb   = (const float*)d_in[5];
    const float* Wo    = (const float*)d_in[6];
    const float* bo    = (const float*)d_in[7];
    float* out = (float*)d_out;

    float* ws = (float*)d_ws;
    const size_t MD = (size_t)kM * kDim;    // 6,291,456 floats
    const size_t MS = (size_t)kM * kDst;    //   262,144 floats
    float* xin = ws;
    float* dlt = ws + MD;
    float* yws = ws + 2 * MD;
    float* Bw  = ws + 3 * MD;
    float* Cw  = Bw + MS;
    // total: 3*MD + 2*MS floats = ~77.6 MB of workspace

    dim3 g1(kM / 64, (kP + 63) / 64);       // 256 x 13
    proj_gemm_kernel<<<g1, 256, 0, stream>>>(x, Wp, bp, dtb, xin, dlt, Bw, Cw);

    scan_kernel<<<kB * (kDim / 16), 256, 0, stream>>>(xin, dlt, Bw, Cw, A_log, Dp, yws);

    dim3 g3(kM / 64, kDim / 64);            // 256 x 6
    out_gemm_kernel<<<g3, 256, 0, stream>>>(yws, Wo, bo, out);
}